// Top2Router_25305947308557
// MI455X (gfx1250) — compile-verified
//
#include <hip/hip_runtime.h>

// ---------------------------------------------------------------------------
// Top-2 MoE router (T=8192, E=64, capacity=320).
// Output (floats, concatenated): used_capacity[E], cb_weight[T*E*C],
// sec_mask[T*E*C], aux_loss, z_loss.  ~1.34 GB, dominated by zero-fill.
// Zero-fill uses the CDNA5 async data mover (LDS -> global, ASYNCcnt).
// ---------------------------------------------------------------------------

#if defined(__HIP_DEVICE_COMPILE__) && defined(__has_builtin)
#if __has_builtin(__builtin_amdgcn_global_store_async_from_lds_b128) && \
    __has_builtin(__builtin_amdgcn_s_wait_asynccnt)
#define ZFILL_ASYNC 1
#endif
#endif

typedef float f4_t __attribute__((ext_vector_type(4)));
typedef int   v4i  __attribute__((ext_vector_type(4)));
typedef __attribute__((address_space(1))) v4i* gv4i_p;  // global int4*
typedef __attribute__((address_space(3))) v4i* lv4i_p;  // LDS int4*

// ------------------------- Kernel A: softmax / top2 -------------------------
// One wave32 per token; lane l owns experts 2l and 2l+1.
__global__ __launch_bounds__(256) void routerA(
    const float* __restrict__ x, int T, int tokensPerBlock,
    int* __restrict__ top1, int* __restrict__ top2,
    float* __restrict__ p1, float* __restrict__ p2,
    float* __restrict__ blkProbs, float* __restrict__ blkZ) {
  __shared__ float sProbs[8 * 64];
  __shared__ float sZ[8];
  const int lane = threadIdx.x & 31;
  const int wave = threadIdx.x >> 5;
  const int tok0 = blockIdx.x * tokensPerBlock;
  int tend = tok0 + tokensPerBlock;
  if (tend > T) tend = T;

  float acc0 = 0.f, acc1 = 0.f, zacc = 0.f;

  for (int t = tok0 + wave; t < tend; t += 8) {
    const float2 v = *(const float2*)(x + (size_t)t * 64 + 2 * lane);
    const float x0 = v.x, x1 = v.y;

    // wave max (softmax stabilization)
    float m = fmaxf(x0, x1);
    for (int o = 16; o; o >>= 1) m = fmaxf(m, __shfl_xor(m, o, 32));
    const float e0 = expf(x0 - m), e1 = expf(x1 - m);
    float se = e0 + e1;
    for (int o = 16; o; o >>= 1) se += __shfl_xor(se, o, 32);
    const float inv = 1.0f / se;

    // argmax (value desc, index asc tie-break == jnp.argmax)
    float bv = x0; int bi = 2 * lane;
    if (x1 > bv) { bv = x1; bi = 2 * lane + 1; }
    float v1v = bv; int v1i = bi;
    for (int o = 16; o; o >>= 1) {
      const float ov = __shfl_xor(v1v, o, 32);
      const int   oi = __shfl_xor(v1i, o, 32);
      if (ov > v1v || (ov == v1v && oi < v1i)) { v1v = ov; v1i = oi; }
    }
    // arg2max: mask out top1 index
    const float c0 = (2 * lane     == v1i) ? -__builtin_inff() : x0;
    const float c1 = (2 * lane + 1 == v1i) ? -__builtin_inff() : x1;
    float v2v = c0; int v2i = 2 * lane;
    if (c1 > v2v) { v2v = c1; v2i = 2 * lane + 1; }
    for (int o = 16; o; o >>= 1) {
      const float ov = __shfl_xor(v2v, o, 32);
      const int   oi = __shfl_xor(v2i, o, 32);
      if (ov > v2v || (ov == v2v && oi < v2i)) { v2v = ov; v2i = oi; }
    }

    acc0 += e0 * inv;
    acc1 += e1 * inv;
    if (lane == 0) {
      top1[t] = v1i;
      top2[t] = v2i;
      p1[t] = expf(v1v - m) * inv;
      p2[t] = expf(v2v - m) * inv;
      const float lse = m + logf(se);
      zacc += lse * lse;
    }
  }

  sProbs[wave * 64 + 2 * lane]     = acc0;
  sProbs[wave * 64 + 2 * lane + 1] = acc1;
  if (lane == 0) sZ[wave] = zacc;
  __syncthreads();
  if (threadIdx.x < 64) {
    float s = 0.f;
    for (int w = 0; w < 8; ++w) s += sProbs[w * 64 + threadIdx.x];
    blkProbs[blockIdx.x * 64 + threadIdx.x] = s;  // deterministic order
  }
  if (threadIdx.x == 0) {
    float s = 0.f;
    for (int w = 0; w < 8; ++w) s += sZ[w];
    blkZ[blockIdx.x] = s;
  }
}

// --------------- Kernel B: sequential ranking (moe_cumsum) -----------------
// One thread per expert; exact order-dependent prefix over tokens.
__global__ __launch_bounds__(64) void routerB(
    const int* __restrict__ top1, const int* __restrict__ top2,
    int T, int cap,
    int* __restrict__ rank1, int* __restrict__ rank2,
    int* __restrict__ m1, int* __restrict__ m2,
    const float* __restrict__ blkProbs, const float* __restrict__ blkZ, int nblk,
    float* __restrict__ usedCap, float* __restrict__ auxOut, float* __restrict__ zOut) {
  const int e = threadIdx.x;  // 0..63

  int c1 = 0;
  for (int t = 0; t < T; ++t) {
    if (top1[t] == e) {
      const int r = c1++;
      if (r < cap) { rank1[t] = r; m1[t] = 1; }
      else         { rank1[t] = 0; m1[t] = 0; }
    }
  }
  int c2 = 0, kept2 = 0;
  for (int t = 0; t < T; ++t) {
    if (top2[t] == e) {
      const int r = c1 + c2;  // offset by UNCLIPPED top1 count (matches ref)
      ++c2;
      if (r < cap) { rank2[t] = r; m2[t] = 1; ++kept2; }
      else         { rank2[t] = 0; m2[t] = 0; }
    }
  }
  usedCap[e] = (float)((c1 < cap ? c1 : cap) + kept2);

  // aux loss: E^2 * mean_e((c1+c2)/T * sum_probs_e/T) = E * sum_e(...)
  float sp = 0.f;
  for (int b = 0; b < nblk; ++b) sp += blkProbs[b * 64 + e];
  __shared__ float sTerm[64];
  sTerm[e] = ((float)(c1 + c2) / (float)T) * (sp / (float)T);
  __syncthreads();
  if (e == 0) {
    float s = 0.f;
    for (int i = 0; i < 64; ++i) s += sTerm[i];
    auxOut[0] = 64.0f * s;
    float z = 0.f;
    for (int b = 0; b < nblk; ++b) z += blkZ[b];
    zOut[0] = z / (float)T;
  }
}

// ---------------- Kernel C: 1.34 GB zero-fill (async LDS path) -------------
__global__ __launch_bounds__(256) void zf_kernel(float* __restrict__ dst, size_t n4) {
#if defined(ZFILL_ASYNC)
  __shared__ __align__(16) float zbuf[128];  // 512 B of zeros, shared per WG
  if (threadIdx.x < 128) zbuf[threadIdx.x] = 0.0f;
  __syncthreads();
  lv4i_p lsrc = (lv4i_p)&zbuf[(threadIdx.x & 31) * 4];  // 16 B per lane
  size_t i = (size_t)blockIdx.x * blockDim.x + threadIdx.x;
  const size_t stride = (size_t)gridDim.x * blockDim.x;
  // CPol TH=NT (1): streaming store, don't retain 1.34 GB in caches.
  // HW stalls issue if ASYNCcnt would overflow, so just stream the stores.
  for (; i + 3 * stride < n4; i += 4 * stride) {
    __builtin_amdgcn_global_store_async_from_lds_b128(
        (gv4i_p)(dst + i * 4), lsrc, 0, 1);
    __builtin_amdgcn_global_store_async_from_lds_b128(
        (gv4i_p)(dst + (i + stride) * 4), lsrc, 0, 1);
    __builtin_amdgcn_global_store_async_from_lds_b128(
        (gv4i_p)(dst + (i + 2 * stride) * 4), lsrc, 0, 1);
    __builtin_amdgcn_global_store_async_from_lds_b128(
        (gv4i_p)(dst + (i + 3 * stride) * 4), lsrc, 0, 1);
  }
  for (; i < n4; i += stride) {
    __builtin_amdgcn_global_store_async_from_lds_b128(
        (gv4i_p)(dst + i * 4), lsrc, 0, 1);
  }
  __builtin_amdgcn_s_wait_asynccnt(0);
#else
  const f4_t z = {0.f, 0.f, 0.f, 0.f};
  size_t i = (size_t)blockIdx.x * blockDim.x + threadIdx.x;
  const size_t stride = (size_t)gridDim.x * blockDim.x;
  f4_t* d4 = (f4_t*)dst;
  for (; i < n4; i += stride) __builtin_nontemporal_store(z, d4 + i);
#endif
}

// ----------------------- Kernel D: sparse scatter --------------------------
__global__ __launch_bounds__(256) void scatterK(
    const int* __restrict__ top1, const int* __restrict__ top2,
    const float* __restrict__ p1, const float* __restrict__ p2,
    const int* __restrict__ rank1, const int* __restrict__ rank2,
    const int* __restrict__ m1, const int* __restrict__ m2,
    int T, int E, int cap,
    float* __restrict__ cb, float* __restrict__ sec) {
  const int t = blockIdx.x * blockDim.x + threadIdx.x;
  if (t >= T) return;
  const size_t rowbase = (size_t)t * (size_t)E * (size_t)cap;
  if (m1[t]) {
    const size_t o = rowbase + (size_t)top1[t] * cap + rank1[t];
    cb[o] = p1[t];
    sec[o] = 1.0f;
  }
  if (m2[t]) {
    const size_t o = rowbase + (size_t)top2[t] * cap + rank2[t];
    cb[o] = p2[t];
    sec[o] = 1.0f;
  }
}

// ---------------------------------------------------------------------------
extern "C" void kernel_launch(void* const* d_in, const int* in_sizes, int n_in,
                              void* d_out, int out_size, void* d_ws, size_t ws_size,
                              hipStream_t stream) {
  const float* x = (const float*)d_in[0];
  const int E = 64;
  const int T = in_sizes[0] / E;
  int cap = (int)((2.0 * 1.25 * (double)T) / (double)E);  // K * CAP_FACTOR * T / E
  cap += cap & 1;
  if (cap < 4) cap = 4;

  const int NBLK_A = 64;
  const int TPB_A = (T + NBLK_A - 1) / NBLK_A;

  // workspace layout
  int*   top1     = (int*)d_ws;
  int*   top2     = top1 + T;
  int*   rank1    = top2 + T;
  int*   rank2    = rank1 + T;
  int*   m1       = rank2 + T;
  int*   m2       = m1 + T;
  float* p1       = (float*)(m2 + T);
  float* p2       = p1 + T;
  float* blkProbs = p2 + T;                  // NBLK_A * 64
  float* blkZ     = blkProbs + NBLK_A * 64;  // NBLK_A

  // output layout (floats, concatenated tuple)
  float* out     = (float*)d_out;
  float* usedCap = out;                      // E
  float* cb      = out + E;                  // T*E*cap
  const size_t big = (size_t)T * (size_t)E * (size_t)cap;
  float* sec     = cb + big;                 // T*E*cap
  float* aux     = sec + big;                // 1
  float* zls     = aux + 1;                  // 1

  routerA<<<NBLK_A, 256, 0, stream>>>(x, T, TPB_A, top1, top2, p1, p2,
                                      blkProbs, blkZ);
  routerB<<<1, 64, 0, stream>>>(top1, top2, T, cap, rank1, rank2, m1, m2,
                                blkProbs, blkZ, NBLK_A, usedCap, aux, zls);
  const size_t n4 = (2 * big) / 4;  // E=64, cap even -> always divisible by 4
  zf_kernel<<<4096, 256, 0, stream>>>(cb, n4);
  scatterK<<<(T + 255) / 256, 256, 0, stream>>>(top1, top2, p1, p2, rank1, rank2,
                                                m1, m2, T, E, cap, cb, sec);
  (void)n_in; (void)out_size; (void)ws_size;
}